// BPR_84653805404606
// MI455X (gfx1250) — compile-verified
//
#include <hip/hip_runtime.h>

// Problem constants (from reference)
#define NUSER 8192
#define NITEM 40000
#define DIM   64
#define NRATE 100
#define TOPK  50
#define NEGV  (-100000000.0f)

// Kernel tiling
#define UPB     16            // users per block (one WMMA M tile)
#define CHUNK   512           // items per chunk staged in LDS
#define NTILE   (CHUNK/16)    // 32 item tiles per chunk
#define TPB     256           // 8 waves of 32
#define NWAVE   (TPB/32)
#define TPW     (NTILE/NWAVE) // tiles per wave per chunk = 4
#define CAP     128           // candidate queue capacity per user per chunk
#define FNEG    (-3.0e38f)
#define TILE_EL (16*DIM)      // 1024 floats = 4 KB per item tile

typedef __attribute__((ext_vector_type(2))) float v2f;
typedef __attribute__((ext_vector_type(8))) float v8f;
typedef unsigned int u32;
typedef __attribute__((ext_vector_type(4))) u32 v4u;
typedef __attribute__((ext_vector_type(8))) int v8i;
typedef __attribute__((ext_vector_type(4))) int v4i;

#if defined(__gfx1250__) && __has_builtin(__builtin_amdgcn_tensor_load_to_lds) && __has_builtin(__builtin_amdgcn_s_wait_tensorcnt)
#define USE_TDM 1
#else
#define USE_TDM 0
#endif

#if USE_TDM
// 1-D TDM copy: 'elems' f32 from gsrc -> LDS byte offset lds_off.
// D# group0: count=1, lds_addr, 57-bit global addr, type=2.
// D# group1: data_size=4B, tensor_dim0=elems, tensor_dim1=1, tile_dim0=elems (1-D tile),
//            tensor_dim0_stride=elems; everything else zero.
// This toolchain's builtin takes 6 args (g0, g1, g2, g3, g4, cpol); g2..g4 unused here.
__device__ __forceinline__ void tdm_load_1d(u32 lds_off, const float* gsrc, u32 elems) {
    unsigned long long ga = (unsigned long long)(uintptr_t)gsrc;
    v4u g0;
    g0.x = 1u;                                               // count=1
    g0.y = lds_off;                                          // lds_addr[31:0]
    g0.z = (u32)(ga & 0xFFFFFFFFu);                          // global_addr[31:0]
    g0.w = (u32)((ga >> 32) & 0x1FFFFFFu) | (2u << 30);      // global_addr[56:32] | type=2
    v8i g1;
    g1[0] = (int)(2u << 16);                                 // data_size=2 (4 bytes)
    g1[1] = (int)((elems & 0xFFFFu) << 16);                  // tensor_dim0[15:0]
    g1[2] = (int)(((elems >> 16) & 0xFFFFu) | (1u << 16));   // tensor_dim0[31:16], tensor_dim1=1 (lo)
    g1[3] = (int)((elems & 0xFFFFu) << 16);                  // tensor_dim1 hi=0, tile_dim0=elems
    g1[4] = 0;                                               // tile_dim1=0 (1-D), tile_dim2=0
    g1[5] = (int)elems;                                      // tensor_dim0_stride[31:0]
    g1[6] = 0;
    g1[7] = 0;
    v4i z4 = {0, 0, 0, 0};
    v8i z8 = {0, 0, 0, 0, 0, 0, 0, 0};
    __builtin_amdgcn_tensor_load_to_lds(g0, g1, z4, z4, z8, 0);
}
#endif

__global__ __launch_bounds__(TPB) void topk_ip_kernel(
    const float* __restrict__ user_embs,   // [NUSER, DIM]
    const float* __restrict__ item_embs,   // [NITEM, DIM]
    const int*   __restrict__ rated,       // [NUSER, NRATE]
    int*         __restrict__ out)         // [NUSER, TOPK]
{
    __shared__ float sc[UPB * CHUNK];             // 32 KB score tile
    __shared__ float userA[UPB * DIM];            // 4 KB user rows
    __shared__ float topV[UPB * TOPK];
    __shared__ int   topI[UPB * TOPK];
    __shared__ float curMin[UPB];
    __shared__ int   minPos[UPB];
    __shared__ int   candCnt[UPB];
    __shared__ int   ovf[UPB];
    __shared__ float candV[UPB * CAP];
    __shared__ int   candI[UPB * CAP];
#if USE_TDM
    __shared__ float stage[NWAVE * 2 * TILE_EL];  // 64 KB: double-buffered tile per wave
#endif

    const int t     = threadIdx.x;
    const int wave  = t >> 5;
    const int lane  = t & 31;
    const int half  = lane >> 4;     // 0: lanes 0-15, 1: lanes 16-31
    const int l16   = lane & 15;
    const int ubase = blockIdx.x * UPB;

    // ---- init shared state ----
    for (int j = t; j < UPB * DIM; j += TPB)
        userA[j] = user_embs[(size_t)(ubase + j / DIM) * DIM + (j % DIM)];
    for (int j = t; j < UPB * TOPK; j += TPB) { topV[j] = FNEG; topI[j] = 0; }
    if (t < UPB) { curMin[t] = FNEG; minPos[t] = 0; candCnt[t] = 0; ovf[t] = 0; }
    __syncthreads();

#if USE_TDM
    float* ws0 = &stage[wave * 2 * TILE_EL];
    const u32 ws_off0 = (u32)(uintptr_t)(void*)ws0;              // LDS byte offset (addr[31:0])
    const u32 ws_off1 = ws_off0 + (u32)(TILE_EL * sizeof(float));
#endif

    const int nChunks = (NITEM + CHUNK - 1) / CHUNK;   // 79
    for (int ch = 0; ch < nChunks; ++ch) {
        const int c0 = ch * CHUNK;

        // ---- GEMM phase: each wave owns TPW 16x16 tiles of this chunk ----
#if USE_TDM
        {
            // prologue: stage first tile for this wave
            int ib0 = c0 + wave * 16;
            if (ib0 > NITEM - 16) ib0 = NITEM - 16;   // clamp (NITEM % 16 == 0 -> tiles never split)
            tdm_load_1d(ws_off0, item_embs + (size_t)ib0 * DIM, TILE_EL);
        }
        for (int it = 0; it < TPW; ++it) {
            const int tl = wave + NWAVE * it;
            const float* cur = (it & 1) ? (ws0 + TILE_EL) : ws0;
            if (it + 1 < TPW) {
                int ibn = c0 + (tl + NWAVE) * 16;
                if (ibn > NITEM - 16) ibn = NITEM - 16;
                tdm_load_1d(((it + 1) & 1) ? ws_off1 : ws_off0,
                            item_embs + (size_t)ibn * DIM, TILE_EL);
                __builtin_amdgcn_s_wait_tensorcnt(1);   // current tile landed
            } else {
                __builtin_amdgcn_s_wait_tensorcnt(0);
            }
            asm volatile("" ::: "memory");              // don't hoist LDS reads above the wait

            const int nG = c0 + tl * 16 + l16;          // global item id for this lane's column
            // A 16x4 f32: lane = M, VGPR r at half h holds K = kb + r, kb = 4*kk + 2h (same for B columns)
            v2f afr[16], bfr[16];
            #pragma unroll
            for (int kk = 0; kk < 16; ++kk) {
                const int kb = kk * 4 + 2 * half;
                afr[kk].x = userA[l16 * DIM + kb];
                afr[kk].y = userA[l16 * DIM + kb + 1];
                bfr[kk].x = cur[l16 * DIM + kb];
                bfr[kk].y = cur[l16 * DIM + kb + 1];
            }
            v8f acc = {};
            #pragma unroll
            for (int kk = 0; kk < 16; ++kk)
                acc = __builtin_amdgcn_wmma_f32_16x16x4_f32(
                    false, afr[kk], false, bfr[kk], (short)0, acc, false, false);

            // C/D layout: VGPR r, half h -> row (r + 8h), column = l16
            #pragma unroll
            for (int r = 0; r < 8; ++r)
                sc[(r + 8 * half) * CHUNK + tl * 16 + l16] = (nG < NITEM) ? acc[r] : FNEG;
            asm volatile("" ::: "memory");              // LDS reads done before next TDM overwrite
        }
#else
        for (int tile = wave; tile < NTILE; tile += NWAVE) {
            const int nG = c0 + tile * 16 + l16;
            const int nC = (nG < NITEM) ? nG : (NITEM - 1);
            const float* brow = item_embs + (size_t)nC * DIM;
            v2f afr[16], bfr[16];
            #pragma unroll
            for (int kk = 0; kk < 16; ++kk) {
                const int kb = kk * 4 + 2 * half;
                afr[kk].x = userA[l16 * DIM + kb];
                afr[kk].y = userA[l16 * DIM + kb + 1];
                bfr[kk] = *(const v2f*)(brow + kb);
            }
            v8f acc = {};
            #pragma unroll
            for (int kk = 0; kk < 16; ++kk)
                acc = __builtin_amdgcn_wmma_f32_16x16x4_f32(
                    false, afr[kk], false, bfr[kk], (short)0, acc, false, false);
            #pragma unroll
            for (int r = 0; r < 8; ++r)
                sc[(r + 8 * half) * CHUNK + tile * 16 + l16] = (nG < NITEM) ? acc[r] : FNEG;
        }
#endif
        __syncthreads();

        // ---- mask rated items that fall in this chunk ----
        for (int j = t; j < UPB * NRATE; j += TPB) {
            const int u  = j / NRATE;
            const int ri = rated[(size_t)(ubase + u) * NRATE + (j % NRATE)];
            const int lc = ri - c0;
            if (lc >= 0 && lc < CHUNK) sc[u * CHUNK + lc] = NEGV;
        }
        __syncthreads();

        // ---- candidate collection: 16 threads per user vs running threshold ----
        {
            const int u = t >> 4;
            const int s = t & 15;
            const float thr = curMin[u];
            for (int c = s; c < CHUNK; c += 16) {
                const float v = sc[u * CHUNK + c];
                if (v > thr) {
                    const int p = atomicAdd(&candCnt[u], 1);
                    if (p < CAP) { candV[u * CAP + p] = v; candI[u * CAP + p] = c0 + c; }
                    else ovf[u] = 1;
                }
            }
        }
        __syncthreads();

        // ---- serial merge into exact top-50 (one thread per user) ----
        if (t < UPB) {
            const int u = t;
            float mn = curMin[u];
            int   mp = minPos[u];
            if (ovf[u]) {
                for (int c = 0; c < CHUNK; ++c) {   // exact fallback: rescan whole row
                    const float v = sc[u * CHUNK + c];
                    if (v > mn) {
                        topV[u * TOPK + mp] = v;
                        topI[u * TOPK + mp] = c0 + c;
                        mn = topV[u * TOPK]; mp = 0;
                        for (int q = 1; q < TOPK; ++q)
                            if (topV[u * TOPK + q] < mn) { mn = topV[u * TOPK + q]; mp = q; }
                    }
                }
            } else {
                const int n = candCnt[u];
                for (int p = 0; p < n; ++p) {
                    const float v = candV[u * CAP + p];
                    if (v > mn) {
                        topV[u * TOPK + mp] = v;
                        topI[u * TOPK + mp] = candI[u * CAP + p];
                        mn = topV[u * TOPK]; mp = 0;
                        for (int q = 1; q < TOPK; ++q)
                            if (topV[u * TOPK + q] < mn) { mn = topV[u * TOPK + q]; mp = q; }
                    }
                }
            }
            curMin[u] = mn; minPos[u] = mp;
            candCnt[u] = 0; ovf[u] = 0;
        }
        __syncthreads();
    }

    // ---- final: sort 50 entries descending (index tie-break low-first) and emit ----
    if (t < UPB) {
        const int u = t;
        for (int i = 0; i < TOPK; ++i) {
            int best = i;
            for (int j = i + 1; j < TOPK; ++j) {
                const float vj = topV[u * TOPK + j];
                const float vb = topV[u * TOPK + best];
                if (vj > vb || (vj == vb && topI[u * TOPK + j] < topI[u * TOPK + best]))
                    best = j;
            }
            const float tv = topV[u * TOPK + i]; const int ti = topI[u * TOPK + i];
            topV[u * TOPK + i] = topV[u * TOPK + best]; topI[u * TOPK + i] = topI[u * TOPK + best];
            topV[u * TOPK + best] = tv; topI[u * TOPK + best] = ti;
            out[(size_t)(ubase + u) * TOPK + i] = topI[u * TOPK + i];
        }
    }
}

extern "C" void kernel_launch(void* const* d_in, const int* in_sizes, int n_in,
                              void* d_out, int out_size, void* d_ws, size_t ws_size,
                              hipStream_t stream) {
    const float* user_embs = (const float*)d_in[0];   // [8192, 64] f32
    const float* item_embs = (const float*)d_in[1];   // [40000, 64] f32
    const int*   rated     = (const int*)d_in[2];     // [8192, 100] i32
    (void)in_sizes; (void)n_in; (void)d_ws; (void)ws_size; (void)out_size;

    int* out = (int*)d_out;                           // [8192, 50] i32 indices

    dim3 grid(NUSER / UPB);   // 512 workgroups
    dim3 block(TPB);          // 8 waves (wave32)
    topk_ip_kernel<<<grid, block, 0, stream>>>(user_embs, item_embs, rated, out);
}